// GraphRCNN_10952166605415
// MI455X (gfx1250) — compile-verified
//
#include <hip/hip_runtime.h>
#include <hip/hip_bf16.h>

typedef __attribute__((ext_vector_type(2))) float v2f;
typedef __attribute__((ext_vector_type(8))) float v8f;

#define N_NODES 256
#define FEAT    1024
#define HID     256
#define HEADS   4
#define OUT_C   21
#define TOPK    3
#define IN_GCN  1028
#define MAX_DEG 800

// ---------------------------------------------------------------------------
// Transpose (and zero-pad) weights: Bt[n, k] = (n < N) ? B[k, n] : 0.
// Bt is Nt x K with row stride K. One-time, L2-resident, ~10 MB total.
// ---------------------------------------------------------------------------
__global__ void transpose_pad_kernel(const float* __restrict__ B,
                                     float* __restrict__ Bt,
                                     int K, int N, int ldb)
{
    const int n = blockIdx.x;
    const int k = blockIdx.y * blockDim.x + threadIdx.x;
    if (k >= K) return;
    float v = 0.f;
    if (n < N) v = B[(size_t)k * ldb + n];
    Bt[(size_t)n * K + k] = v;
}

// ---------------------------------------------------------------------------
// f32 WMMA GEMM with pre-transposed B: C[M x N] = A[M x K] * Bt[N x K]^T
// via V_WMMA_F32_16X16X4_F32. One wave computes a 16x64 tile with 4
// independent accumulator chains. Software-pipelined: the next K-step's 5
// b64 fragments live in separate registers from the current set, so their
// loads overlap the 4 WMMAs of the current step instead of each WMMA doing
// a full s_wait_loadcnt 0.  Ragged N handled by zero-padded Bt rows; the
// inner loop is fully branchless. blockDim = (32, 4).
// ---------------------------------------------------------------------------
__global__ void wmma_gemm_tb_kernel(const float* __restrict__ A,
                                    const float* __restrict__ Bt,
                                    float* __restrict__ C,
                                    int K, int lda, int ldt, int ldc,
                                    int Npad, int Nreal)
{
    const int lane = threadIdx.x;                       // 0..31
    const int grp  = blockIdx.x * blockDim.y + threadIdx.y;
    if (grp * 64 >= Npad) return;
    const int m0   = blockIdx.y * 16;
    const int n0   = grp * 64;
    const int half = lane >> 4;                         // 0: K=0,1  1: K=2,3
    const int l    = lane & 15;

    const v2f* ap  = (const v2f*)(A  + (size_t)(m0 + l) * lda + 2 * half);
    const v2f* bp0 = (const v2f*)(Bt + (size_t)(n0 +  0 + l) * ldt + 2 * half);
    const v2f* bp1 = (const v2f*)(Bt + (size_t)(n0 + 16 + l) * ldt + 2 * half);
    const v2f* bp2 = (const v2f*)(Bt + (size_t)(n0 + 32 + l) * ldt + 2 * half);
    const v2f* bp3 = (const v2f*)(Bt + (size_t)(n0 + 48 + l) * ldt + 2 * half);

    // Warm the caches for the first chunk of each stream (global_prefetch_b8).
    __builtin_prefetch(ap,  0, 1);
    __builtin_prefetch(bp0, 0, 1);
    __builtin_prefetch(bp1, 0, 1);
    __builtin_prefetch(bp2, 0, 1);
    __builtin_prefetch(bp3, 0, 1);

    v8f acc0 = {}, acc1 = {}, acc2 = {}, acc3 = {};

    // Prologue: fragments for K-step 0.
    v2f ac  = ap[0];
    v2f b0c = bp0[0], b1c = bp1[0], b2c = bp2[0], b3c = bp3[0];

    for (int k = 0; k + 4 < K; k += 4) {
        // Issue next step's loads (independent of this step's WMMAs).
        const v2f an  = ap[2];
        const v2f b0n = bp0[2];
        const v2f b1n = bp1[2];
        const v2f b2n = bp2[2];
        const v2f b3n = bp3[2];
        ap += 2; bp0 += 2; bp1 += 2; bp2 += 2; bp3 += 2;

        acc0 = __builtin_amdgcn_wmma_f32_16x16x4_f32(false, ac, false, b0c, (short)0, acc0, false, false);
        acc1 = __builtin_amdgcn_wmma_f32_16x16x4_f32(false, ac, false, b1c, (short)0, acc1, false, false);
        acc2 = __builtin_amdgcn_wmma_f32_16x16x4_f32(false, ac, false, b2c, (short)0, acc2, false, false);
        acc3 = __builtin_amdgcn_wmma_f32_16x16x4_f32(false, ac, false, b3c, (short)0, acc3, false, false);

        ac = an; b0c = b0n; b1c = b1n; b2c = b2n; b3c = b3n;
    }
    // Epilogue: last K-step.
    acc0 = __builtin_amdgcn_wmma_f32_16x16x4_f32(false, ac, false, b0c, (short)0, acc0, false, false);
    acc1 = __builtin_amdgcn_wmma_f32_16x16x4_f32(false, ac, false, b1c, (short)0, acc1, false, false);
    acc2 = __builtin_amdgcn_wmma_f32_16x16x4_f32(false, ac, false, b2c, (short)0, acc2, false, false);
    acc3 = __builtin_amdgcn_wmma_f32_16x16x4_f32(false, ac, false, b3c, (short)0, acc3, false, false);

    // C layout per sub-tile: VGPR r -> (M = m0 + r + 8*half, N = n0 + j*16 + l)
    const int c0 = n0 + l;
    #pragma unroll
    for (int r = 0; r < 8; ++r) {
        const int m = m0 + r + half * 8;
        float* crow = C + (size_t)m * ldc;
        if (c0      < Nreal) crow[c0]      = acc0[r];
        if (c0 + 16 < Nreal) crow[c0 + 16] = acc1[r];
        if (c0 + 32 < Nreal) crow[c0 + 32] = acc2[r];
        if (c0 + 48 < Nreal) crow[c0 + 48] = acc3[r];
    }
}

// ---------------------------------------------------------------------------
// Relation scores (factored pair-MLP) fused with per-row top-4 selection.
// Block = one row i (256 threads, thread j = column j).
// ---------------------------------------------------------------------------
__global__ void rel_topk_kernel(const float* __restrict__ Aproj,
                                const float* __restrict__ Bproj,
                                const float* __restrict__ boxes,
                                const float* __restrict__ Wfc1,
                                const float* __restrict__ bfc1,
                                const float* __restrict__ Wfc2,
                                const float* __restrict__ bfc2,
                                int* __restrict__ tgt)
{
    __shared__ float sA[HID];
    __shared__ float sG[4][HID];
    __shared__ float sV[HID];
    __shared__ float vals[N_NODES];
    const int i = blockIdx.x;
    const int j = threadIdx.x;

    sA[j] = Aproj[i * HID + j] + bfc1[j];
    #pragma unroll
    for (int k = 0; k < 4; ++k) sG[k][j] = Wfc1[(size_t)(2048 + k) * HID + j];
    sV[j] = Wfc2[j];
    __syncthreads();

    const float d0 = fabsf(boxes[i * 4 + 0] - boxes[j * 4 + 0]);
    const float d1 = fabsf(boxes[i * 4 + 1] - boxes[j * 4 + 1]);
    const float d2 = fabsf(boxes[i * 4 + 2] - boxes[j * 4 + 2]);
    const float d3 = fabsf(boxes[i * 4 + 3] - boxes[j * 4 + 3]);
    const float* Brow = Bproj + (size_t)j * HID;

    float acc = 0.f;
    for (int c = 0; c < HID; ++c) {
        float v = sA[c] + Brow[c] + d0 * sG[0][c] + d1 * sG[1][c]
                                  + d2 * sG[2][c] + d3 * sG[3][c];
        acc += fmaxf(v, 0.f) * sV[c];
    }
    vals[j] = acc + bfc2[0];
    __syncthreads();

    // top-4 (jax.lax.top_k tie-break: smallest index first); drop the first.
    if (j == 0) {
        unsigned used[8] = {0, 0, 0, 0, 0, 0, 0, 0};
        for (int t = 0; t < TOPK + 1; ++t) {
            float best = -3.0e38f; int bi = 0;
            for (int q = 0; q < N_NODES; ++q) {
                if ((used[q >> 5] >> (q & 31)) & 1u) continue;
                if (vals[q] > best) { best = vals[q]; bi = q; }
            }
            used[bi >> 5] |= 1u << (bi & 31);
            if (t >= 1) tgt[i * TOPK + (t - 1)] = bi;
        }
    }
}

// ---------------------------------------------------------------------------
// Build gcn_in = [features | normalized box geometry]   (256 x 1028)
// ---------------------------------------------------------------------------
__global__ void gcn_in_kernel(const float* __restrict__ features,
                              const float* __restrict__ boxes,
                              const int* __restrict__ img_h,
                              const int* __restrict__ img_w,
                              float* __restrict__ gcn_in)
{
    const int n = blockIdx.x;
    const int t = threadIdx.x;  // 256
    #pragma unroll
    for (int r = 0; r < 4; ++r)
        gcn_in[(size_t)n * IN_GCN + r * 256 + t] =
            features[(size_t)n * FEAT + r * 256 + t];
    if (t == 0) {
        const float w = (float)img_w[0], h = (float)img_h[0];
        const float x1 = boxes[n * 4 + 0] / w, y1 = boxes[n * 4 + 1] / h;
        const float x2 = boxes[n * 4 + 2] / w, y2 = boxes[n * 4 + 3] / h;
        gcn_in[(size_t)n * IN_GCN + 1024] = x1;
        gcn_in[(size_t)n * IN_GCN + 1025] = y1;
        gcn_in[(size_t)n * IN_GCN + 1026] = x2 - x1;
        gcn_in[(size_t)n * IN_GCN + 1027] = y2 - y1;
    }
}

// ---------------------------------------------------------------------------
// Per-node attention coefficients, layer 1: as1/ad1[n,h] = h[n,h,:]·a_{src,dst}
// ---------------------------------------------------------------------------
__global__ void attn_coef1_kernel(const float* __restrict__ H1,
                                  const float* __restrict__ a_src,
                                  const float* __restrict__ a_dst,
                                  float* __restrict__ as1,
                                  float* __restrict__ ad1)
{
    __shared__ float red[256];
    const int n = blockIdx.x, t = threadIdx.x;
    for (int h = 0; h < HEADS; ++h) {
        const float hv = H1[(size_t)n * 1024 + h * 256 + t];
        red[t] = hv * a_src[h * 256 + t];
        __syncthreads();
        for (int s = 128; s > 0; s >>= 1) { if (t < s) red[t] += red[t + s]; __syncthreads(); }
        if (t == 0) as1[n * HEADS + h] = red[0];
        __syncthreads();
        red[t] = hv * a_dst[h * 256 + t];
        __syncthreads();
        for (int s = 128; s > 0; s >>= 1) { if (t < s) red[t] += red[t + s]; __syncthreads(); }
        if (t == 0) ad1[n * HEADS + h] = red[0];
        __syncthreads();
    }
}

// ---------------------------------------------------------------------------
// GAT layer-1 aggregation (softmax over incoming edges) + bias + ReLU.
// Block = one target node t. Edge gather is serial (deterministic order).
// ---------------------------------------------------------------------------
__global__ void gat1_agg_kernel(const int* __restrict__ tgt,
                                const float* __restrict__ H1,
                                const float* __restrict__ as1,
                                const float* __restrict__ ad1,
                                const float* __restrict__ b1,
                                float* __restrict__ h1out)
{
    __shared__ int   list[MAX_DEG];
    __shared__ float alpha[MAX_DEG * HEADS];
    __shared__ int   s_deg;
    const int t = blockIdx.x;
    const int tid = threadIdx.x;

    if (tid == 0) {
        int deg = 0;
        for (int e = 0; e < N_NODES * TOPK; ++e)
            if (tgt[e] == t && deg < MAX_DEG - 1) list[deg++] = e / TOPK;
        list[deg++] = t;                         // GATConv self loop
        s_deg = deg;
    }
    __syncthreads();
    const int deg = s_deg;

    if (tid < HEADS) {
        const int h = tid;
        const float adt = ad1[t * HEADS + h];
        float m = -3.0e38f;
        for (int k = 0; k < deg; ++k) {
            float e = as1[list[k] * HEADS + h] + adt;
            e = e > 0.f ? e : 0.2f * e;          // leaky_relu(0.2)
            alpha[k * HEADS + h] = e;
            m = fmaxf(m, e);
        }
        float den = 0.f;
        for (int k = 0; k < deg; ++k) {
            float p = expf(alpha[k * HEADS + h] - m);
            alpha[k * HEADS + h] = p; den += p;
        }
        const float inv = 1.f / den;
        for (int k = 0; k < deg; ++k) alpha[k * HEADS + h] *= inv;
    }
    __syncthreads();

    const int c = tid;                            // 256 threads: channel c per head
    float a0 = 0.f, a1 = 0.f, a2 = 0.f, a3 = 0.f;
    for (int k = 0; k < deg; ++k) {
        const float* Hs = H1 + (size_t)list[k] * 1024;
        a0 += alpha[k * HEADS + 0] * Hs[c];
        a1 += alpha[k * HEADS + 1] * Hs[256 + c];
        a2 += alpha[k * HEADS + 2] * Hs[512 + c];
        a3 += alpha[k * HEADS + 3] * Hs[768 + c];
    }
    h1out[(size_t)t * 1024 + c]       = fmaxf(a0 + b1[c], 0.f);
    h1out[(size_t)t * 1024 + 256 + c] = fmaxf(a1 + b1[256 + c], 0.f);
    h1out[(size_t)t * 1024 + 512 + c] = fmaxf(a2 + b1[512 + c], 0.f);
    h1out[(size_t)t * 1024 + 768 + c] = fmaxf(a3 + b1[768 + c], 0.f);
}

// ---------------------------------------------------------------------------
// Per-node attention coefficients, layer 2 (1 head, 21 channels)
// ---------------------------------------------------------------------------
__global__ void attn_coef2_kernel(const float* __restrict__ H2,
                                  const float* __restrict__ a_src2,
                                  const float* __restrict__ a_dst2,
                                  float* __restrict__ as2,
                                  float* __restrict__ ad2)
{
    const int n = threadIdx.x;
    if (n >= N_NODES) return;
    float s = 0.f, d = 0.f;
    for (int o = 0; o < OUT_C; ++o) {
        const float hv = H2[n * OUT_C + o];
        s += hv * a_src2[o];
        d += hv * a_dst2[o];
    }
    as2[n] = s; ad2[n] = d;
}

// ---------------------------------------------------------------------------
// GAT layer-2 aggregation + bias -> logits, plus argmax -> labels.
// Block = one target node (32 threads).
// ---------------------------------------------------------------------------
__global__ void gat2_out_kernel(const int* __restrict__ tgt,
                                const float* __restrict__ H2,
                                const float* __restrict__ as2,
                                const float* __restrict__ ad2,
                                const float* __restrict__ b2,
                                float* __restrict__ out)
{
    __shared__ int   list[MAX_DEG];
    __shared__ float alpha[MAX_DEG];
    __shared__ float lg[OUT_C];
    __shared__ int   s_deg;
    const int t = blockIdx.x, tid = threadIdx.x;

    if (tid == 0) {
        int deg = 0;
        for (int e = 0; e < N_NODES * TOPK; ++e)
            if (tgt[e] == t && deg < MAX_DEG - 1) list[deg++] = e / TOPK;
        list[deg++] = t;
        const float adt = ad2[t];
        float m = -3.0e38f;
        for (int k = 0; k < deg; ++k) {
            float e = as2[list[k]] + adt;
            e = e > 0.f ? e : 0.2f * e;
            alpha[k] = e; m = fmaxf(m, e);
        }
        float den = 0.f;
        for (int k = 0; k < deg; ++k) { float p = expf(alpha[k] - m); alpha[k] = p; den += p; }
        const float inv = 1.f / den;
        for (int k = 0; k < deg; ++k) alpha[k] *= inv;
        s_deg = deg;
    }
    __syncthreads();
    const int deg = s_deg;

    if (tid < OUT_C) {
        float acc = 0.f;
        for (int k = 0; k < deg; ++k) acc += alpha[k] * H2[list[k] * OUT_C + tid];
        const float v = acc + b2[tid];
        lg[tid] = v;
        out[t * OUT_C + tid] = v;
    }
    __syncthreads();
    if (tid == 0) {
        int best = 0; float bv = lg[0];
        for (int o = 1; o < OUT_C; ++o) if (lg[o] > bv) { bv = lg[o]; best = o; }
        out[N_NODES * OUT_C + t] = (float)best;   // labels after logits
    }
}

// ---------------------------------------------------------------------------
extern "C" void kernel_launch(void* const* d_in, const int* in_sizes, int n_in,
                              void* d_out, int out_size, void* d_ws, size_t ws_size,
                              hipStream_t stream)
{
    (void)in_sizes; (void)n_in; (void)out_size; (void)ws_size;
    const float* features = (const float*)d_in[0];
    const float* boxes    = (const float*)d_in[1];
    const float* W_fc1    = (const float*)d_in[2];
    const float* b_fc1    = (const float*)d_in[3];
    const float* W_fc2    = (const float*)d_in[4];
    const float* b_fc2    = (const float*)d_in[5];
    const float* W1       = (const float*)d_in[6];
    const float* a_src1   = (const float*)d_in[7];
    const float* a_dst1   = (const float*)d_in[8];
    const float* b1       = (const float*)d_in[9];
    const float* W2       = (const float*)d_in[10];
    const float* a_src2   = (const float*)d_in[11];
    const float* a_dst2   = (const float*)d_in[12];
    const float* b2       = (const float*)d_in[13];
    const int*   img_h    = (const int*)d_in[14];
    const int*   img_w    = (const int*)d_in[15];

    float* ws    = (float*)d_ws;
    float* Aproj = ws;                  // 256*256
    float* Bproj = Aproj + 65536;       // 256*256
    float* gcnin = Bproj + 65536;       // 256*1028
    float* H1    = gcnin + 263168;      // 256*1024
    float* as1   = H1 + 262144;         // 256*4
    float* ad1   = as1 + 1024;          // 256*4
    float* h1    = ad1 + 1024;          // 256*1024
    float* H2    = h1 + 262144;         // 256*21
    float* as2   = H2 + 5376;           // 256
    float* ad2   = as2 + 256;           // 256
    int*   tgt   = (int*)(ad2 + 256);   // 768 ints
    float* WtA   = (float*)(tgt + 768); // 256*1024  (W_fc1[0:1024]^T)
    float* WtB   = WtA + 262144;        // 256*1024  (W_fc1[1024:2048]^T)
    float* W1t   = WtB + 262144;        // 1024*1028 (W1^T)
    float* W2t   = W1t + 1052672;       // 64*1024   (W2^T, rows>=21 zeroed)
    float* out   = (float*)d_out;

    // One-time weight transposes (zero-padded where N is ragged).
    transpose_pad_kernel<<<dim3(256, 4),  256, 0, stream>>>(W_fc1,              WtA, 1024, 256,  256);
    transpose_pad_kernel<<<dim3(256, 4),  256, 0, stream>>>(W_fc1 + 1024 * 256, WtB, 1024, 256,  256);
    transpose_pad_kernel<<<dim3(1024, 5), 256, 0, stream>>>(W1,                 W1t, 1028, 1024, 1024);
    transpose_pad_kernel<<<dim3(64, 4),   256, 0, stream>>>(W2,                 W2t, 1024, 21,   21);

    dim3 wblk(32, 4);
    // Aproj = F @ W_fc1[0:1024,:], Bproj = F @ W_fc1[1024:2048,:]
    wmma_gemm_tb_kernel<<<dim3(1, 16), wblk, 0, stream>>>(
        features, WtA, Aproj, 1024, 1024, 1024, 256, 256, 256);
    wmma_gemm_tb_kernel<<<dim3(1, 16), wblk, 0, stream>>>(
        features, WtB, Bproj, 1024, 1024, 1024, 256, 256, 256);
    gcn_in_kernel<<<256, 256, 0, stream>>>(features, boxes, img_h, img_w, gcnin);
    // H1 = gcn_in @ W1   (256 x 1028 x 1024): 16 x 64-col groups
    wmma_gemm_tb_kernel<<<dim3(4, 16), wblk, 0, stream>>>(
        gcnin, W1t, H1, 1028, 1028, 1028, 1024, 1024, 1024);
    rel_topk_kernel<<<256, 256, 0, stream>>>(
        Aproj, Bproj, boxes, W_fc1, b_fc1, W_fc2, b_fc2, tgt);
    attn_coef1_kernel<<<256, 256, 0, stream>>>(H1, a_src1, a_dst1, as1, ad1);
    gat1_agg_kernel<<<256, 256, 0, stream>>>(tgt, H1, as1, ad1, b1, h1);
    // H2 = h1 @ W2   (256 x 1024 x 21, N padded to 64 -> 1 group)
    wmma_gemm_tb_kernel<<<dim3(1, 16), wblk, 0, stream>>>(
        h1, W2t, H2, 1024, 1024, 1024, 21, 64, 21);
    attn_coef2_kernel<<<1, 256, 0, stream>>>(H2, a_src2, a_dst2, as2, ad2);
    gat2_out_kernel<<<256, 32, 0, stream>>>(tgt, H2, as2, ad2, b2, out);
}